// WaterNet1_76373108457551
// MI455X (gfx1250) — compile-verified
//
#include <hip/hip_runtime.h>
#include <math.h>

typedef float v2f __attribute__((ext_vector_type(2)));
typedef float v8f __attribute__((ext_vector_type(8)));

#define NH 64
#define NG 32
#define NJ 256   // 4*NH

// ---------------------------------------------------------------------------
// Kernel 1: w = xc @ W3.T + b3  via V_WMMA_F32_16X16X4_F32 (full f32 precision,
// required since outputs feed exp/softmax). One wave32 per 16x16 output tile,
// 8 chained k=4 WMMAs covering ng=32.
// A fragment (16x4 f32): lane<16 -> row=lane, K=k0..k0+1 in v[0..1];
//                        lane>=16 -> row=lane-16, K=k0+2..k0+3.
// B fragment (4x16 f32): same striping with N across lanes.
// C/D (16x16 f32, 8 VGPRs): VGPR r -> row r + 8*(lane>=16), col lane%16.
// ---------------------------------------------------------------------------
__global__ void wn_gates_gemm(const float* __restrict__ xc,
                              const float* __restrict__ W3,
                              const float* __restrict__ b3,
                              float* __restrict__ w_buf,
                              int ns) {
  const int lane = threadIdx.x;          // blockDim.x == 32, EXEC all ones
  const int tilesN = NJ / 16;            // 16
  const int mt  = blockIdx.x / tilesN;
  const int ntl = blockIdx.x % tilesN;
  const int s0 = mt * 16;
  const int j0 = ntl * 16;
  if (s0 >= ns) return;

  const int half = lane >> 4;            // 0: K pair (k0,k0+1); 1: (k0+2,k0+3)
  const int l16  = lane & 15;

  const float* arow = xc + (size_t)(s0 + l16) * NG;   // xc[s][g], row-major
  const float* brow = W3 + (size_t)(j0 + l16) * NG;   // W3[j][g]; B[k][n]=W3[j0+n][k]

  v8f acc = {0.f, 0.f, 0.f, 0.f, 0.f, 0.f, 0.f, 0.f};
#pragma unroll
  for (int k0 = 0; k0 < NG; k0 += 4) {
    const int ka = k0 + half * 2;
    v2f a, b;
    a.x = arow[ka];
    a.y = arow[ka + 1];
    b.x = brow[ka];
    b.y = brow[ka + 1];
    acc = __builtin_amdgcn_wmma_f32_16x16x4_f32(
        /*neg_a=*/false, a, /*neg_b=*/false, b,
        /*c_mod=*/(short)0, acc, /*reuse_a=*/false, /*reuse_b=*/false);
  }

  const float bias = b3[j0 + l16];
  float* out = w_buf + (size_t)(s0 + half * 8) * NJ + j0 + l16;
#pragma unroll
  for (int r = 0; r < 8; ++r) {
    out[(size_t)r * NJ] = acc[r] + bias;
  }
}

// ---------------------------------------------------------------------------
// Kernel 2: gate nonlinearities. One wave32 per catchment s; lane owns hidden
// units h=lane and h=lane+32. Softmax over nh=64 via wave32 shfl reductions.
// ---------------------------------------------------------------------------
__global__ void wn_gates_act(const float* __restrict__ w_buf,
                             float* __restrict__ gm, float* __restrict__ ge,
                             float* __restrict__ go, float* __restrict__ ga,
                             int ns) {
  const int lane = threadIdx.x & 31;
  const int wave = threadIdx.x >> 5;
  const int s = blockIdx.x * (blockDim.x >> 5) + wave;
  if (s >= ns) return;
  const float* w = w_buf + (size_t)s * NJ;
  float* gmo = gm + (size_t)s * NH;
  float* geo = ge + (size_t)s * NH;
  float* goo = go + (size_t)s * NH;
  float* gao = ga + (size_t)s * NH;

  gmo[lane]      = expf(w[lane])      + 1.0f;
  gmo[lane + 32] = expf(w[lane + 32]) + 1.0f;
  geo[lane]      = 1.0f / (1.0f + expf(-w[NH + lane]));
  geo[lane + 32] = 1.0f / (1.0f + expf(-w[NH + lane + 32]));
  goo[lane]      = 1.0f / (1.0f + expf(-w[2 * NH + lane]));
  goo[lane + 32] = 1.0f / (1.0f + expf(-w[2 * NH + lane + 32]));

  // softmax over the 64 attention logits
  float a0 = w[3 * NH + lane];
  float a1 = w[3 * NH + lane + 32];
  float mx = fmaxf(a0, a1);
#pragma unroll
  for (int m = 16; m >= 1; m >>= 1) mx = fmaxf(mx, __shfl_xor(mx, m, 32));
  float e0 = expf(a0 - mx);
  float e1 = expf(a1 - mx);
  float sum = e0 + e1;
#pragma unroll
  for (int m = 16; m >= 1; m >>= 1) sum += __shfl_xor(sum, m, 32);
  const float inv = 1.0f / sum;
  gao[lane]      = e0 * inv;
  gao[lane + 32] = e1 * inv;
}

// ---------------------------------------------------------------------------
// Kernel 3: the time scan. One wave32 per catchment; all state in registers.
// Per step: broadcast float4 forcing load, rain/snow partition (uniform per
// wave, so the branches are convergent), register state update, wave-reduce Y.
// ---------------------------------------------------------------------------
__global__ void wn_scan(const float* __restrict__ x,
                        const float* __restrict__ gm, const float* __restrict__ ge,
                        const float* __restrict__ go, const float* __restrict__ ga,
                        float* __restrict__ yout, int nt, int ns) {
  const int lane = threadIdx.x & 31;
  const int wave = threadIdx.x >> 5;
  const int s = blockIdx.x * (blockDim.x >> 5) + wave;
  if (s >= ns) return;

  const float gm0 = gm[(size_t)s * NH + lane], gm1 = gm[(size_t)s * NH + lane + 32];
  const float ge0 = ge[(size_t)s * NH + lane], ge1 = ge[(size_t)s * NH + lane + 32];
  const float go0 = go[(size_t)s * NH + lane], go1 = go[(size_t)s * NH + lane + 32];
  const float ga0 = ga[(size_t)s * NH + lane], ga1 = ga[(size_t)s * NH + lane + 32];

  float S0 = 0.f, S1 = 0.f, H0 = 0.f, H1 = 0.f;
  const float invPI = 1.0f / 3.1415f;   // module hard-codes 3.1415
  const float4* xv = (const float4*)x;

  for (int t = 0; t < nt; ++t) {
    const float4 v = xv[(size_t)t * ns + s];   // P, E, T1, T2 (wave-broadcast)
    const float P = v.x, E = v.y, T1 = v.z, T2 = v.w;
    const float Ta = 0.5f * (T1 + T2);
    float rP;
    if (T1 >= 0.0f) {
      rP = 1.0f;                               // melted
    } else if (T2 <= 0.0f) {
      rP = 0.0f;                               // frozen
    } else {
      float arg = (T1 + T2) / (T2 - T1);
      arg = fminf(1.0f, fmaxf(-1.0f, arg));
      rP = 1.0f - acosf(arg) * invPI;
    }
    const float Ps = (1.0f - rP) * P;
    const float Pl = rP * P;

    // hidden unit h = lane
    const float melt0 = fmaxf(Ta * gm0, 0.0f);
    const float Sm0   = fminf(S0, melt0);
    const float Hn0   = fmaxf(H0 + Sm0 + Pl - E * ge0, 0.0f);
    const float Q0    = Hn0 * go0;
    S0 = S0 + Ps - Sm0;
    H0 = Hn0 - Q0;

    // hidden unit h = lane + 32
    const float melt1 = fmaxf(Ta * gm1, 0.0f);
    const float Sm1   = fminf(S1, melt1);
    const float Hn1   = fmaxf(H1 + Sm1 + Pl - E * ge1, 0.0f);
    const float Q1    = Hn1 * go1;
    S1 = S1 + Ps - Sm1;
    H1 = Hn1 - Q1;

    float y = Q0 * ga0 + Q1 * ga1;
#pragma unroll
    for (int m = 16; m >= 1; m >>= 1) y += __shfl_xor(y, m, 32);
    if (lane == 0) yout[(size_t)t * ns + s] = y;
  }
}

// ---------------------------------------------------------------------------
// Launcher. Inputs (setup_inputs order): x[nt,ns,4], xc[ns,ng], W3[4nh,ng],
// b3[4nh]. Output: Yout[nt,ns] float32.
// ---------------------------------------------------------------------------
extern "C" void kernel_launch(void* const* d_in, const int* in_sizes, int n_in,
                              void* d_out, int out_size, void* d_ws, size_t ws_size,
                              hipStream_t stream) {
  (void)n_in; (void)out_size; (void)ws_size;
  const float* x  = (const float*)d_in[0];
  const float* xc = (const float*)d_in[1];
  const float* W3 = (const float*)d_in[2];
  const float* b3 = (const float*)d_in[3];
  float* yout = (float*)d_out;

  const int nh4 = in_sizes[3];              // 256
  const int ng  = in_sizes[2] / nh4;        // 32
  const int ns  = in_sizes[1] / ng;         // 1024
  const int nt  = in_sizes[0] / (4 * ns);   // 1000

  float* w_buf = (float*)d_ws;              // [ns, 256]
  float* gm = w_buf + (size_t)ns * NJ;      // [ns, 64] each
  float* ge = gm + (size_t)ns * NH;
  float* go = ge + (size_t)ns * NH;
  float* ga = go + (size_t)ns * NH;

  // 1) gates GEMM: one wave per 16x16 tile
  const int tiles = (ns / 16) * (NJ / 16);  // 1024
  wn_gates_gemm<<<tiles, 32, 0, stream>>>(xc, W3, b3, w_buf, ns);

  // 2) gate nonlinearities: 8 waves/block, one wave per catchment
  wn_gates_act<<<(ns + 7) / 8, 256, 0, stream>>>(w_buf, gm, ge, go, ga, ns);

  // 3) time scan: one wave per catchment, register-resident state
  wn_scan<<<(ns + 7) / 8, 256, 0, stream>>>(x, gm, ge, go, ga, yout, nt, ns);
}